// bayesian_nn_20985210208953
// MI455X (gfx1250) — compile-verified
//
#include <hip/hip_runtime.h>
#include <stdint.h>

// ---------------------------------------------------------------------------
// Bayesian NN MC-sampled forward, MI455X (gfx1250).
// ~362 GFLOP f32 total, ~16MB working set (L2-resident), HBM-trivial
// => matrix-compute bound => V_WMMA_F32_16X16X4_F32 for all 512^3 GEMMs.
// GEMM: 256 threads (8 waves), workgroup tile 64(M)x128(N), 2x2 16x16 tiles
// per wave, K staged in 32-chunks through LDS. B kept K-paired (float2) in
// LDS so each fragment is one ds_load_b64 into an aligned VGPR pair (no movs).
// ---------------------------------------------------------------------------

typedef __attribute__((ext_vector_type(2))) float v2f;
typedef __attribute__((ext_vector_type(8))) float v8f;

// -------------------------- Philox-4x32-10 RNG ------------------------------
__device__ __forceinline__ void philox4x32(uint32_t c0, uint32_t c1, uint32_t c2,
                                           uint32_t c3, uint32_t k0, uint32_t k1,
                                           uint32_t out[4]) {
#pragma unroll
  for (int i = 0; i < 10; ++i) {
    const uint32_t M0 = 0xD2511F53u, M1 = 0xCD9E8D57u;
    uint32_t hi0 = __umulhi(M0, c0), lo0 = M0 * c0;
    uint32_t hi1 = __umulhi(M1, c2), lo1 = M1 * c2;
    uint32_t n0 = hi1 ^ c1 ^ k0;
    uint32_t n1 = lo1;
    uint32_t n2 = hi0 ^ c3 ^ k1;
    uint32_t n3 = lo0;
    c0 = n0; c1 = n1; c2 = n2; c3 = n3;
    k0 += 0x9E3779B9u; k1 += 0xBB67AE85u;
  }
  out[0] = c0; out[1] = c1; out[2] = c2; out[3] = c3;
}

__device__ __forceinline__ void boxmuller(uint32_t u0, uint32_t u1, float& n0, float& n1) {
  float a = ((float)u0 + 0.5f) * 2.3283064365386963e-10f;  // (0,1)
  float b = ((float)u1 + 0.5f) * 2.3283064365386963e-10f;
  float rr = sqrtf(-2.0f * __logf(a));
  float s, c;
  __sincosf(6.283185307179586f * b, &s, &c);
  n0 = rr * c;
  n1 = rr * s;
}

// r[i] = normal * exp(v[i])   (sqrt(exp(2v)) == exp(v)); 512x512 per sample.
__global__ __launch_bounds__(256) void rng_fill(float* __restrict__ out, long long sOut,
                                                const float* __restrict__ v,
                                                int s0, int tensorId) {
  const int tid = threadIdx.x;
  const int s = blockIdx.y;
  float* o = out + (long long)s * sOut;
  const int i4 = (blockIdx.x * 256 + tid) * 4;
  uint32_t r[4];
  philox4x32((uint32_t)(i4 >> 2), (uint32_t)(s0 + s), (uint32_t)tensorId, 0x5EED5EEDu,
             (uint32_t)(s0 + s), 0xB1A5F00Du ^ (uint32_t)tensorId, r);
  float n0, n1, n2, n3;
  boxmuller(r[0], r[1], n0, n1);
  boxmuller(r[2], r[3], n2, n3);
  const float4 vv = *reinterpret_cast<const float4*>(v + i4);
  float4 res;
  res.x = n0 * __expf(vv.x);
  res.y = n1 * __expf(vv.y);
  res.z = n2 * __expf(vv.z);
  res.w = n3 * __expf(vv.w);
  *reinterpret_cast<float4*>(o + i4) = res;
}

// ---------------- 512x512x512 f32 WMMA GEMM, C = A * op(B) ------------------
// TRANSB=0: C = A @ B ; TRANSB=1: C = A @ B^T
// EPI: 0 none; 1 C += E[row,col] (mu matrix); 2 C = relu(C + E[col]) (bias)
template <bool TRANSB, int EPI>
__global__ __launch_bounds__(256) void gemm512(const float* __restrict__ A, long long sA,
                                               const float* __restrict__ B, long long sB,
                                               const float* __restrict__ E, long long sE,
                                               float* __restrict__ C, long long sC) {
  constexpr int K = 512, N = 512;
  __shared__ float As[64][36];     // [m][k]; 36-pad: 16B rows, conflict-free frags
  __shared__ float2 Bs[16][132];   // [k/2][n] K-paired; frag = one b64 load
  const int tid = threadIdx.x;
  const int z = blockIdx.z;
  A += (long long)z * sA;
  B += (long long)z * sB;
  C += (long long)z * sC;
  if (EPI) E += (long long)z * sE;
  const int m0 = blockIdx.y * 64, n0 = blockIdx.x * 128;
  const int wave = tid >> 5, lid = tid & 31;
  const int half = lid >> 4, lm = lid & 15;
  const int wm = wave >> 2;              // 0..1 : 32-row band of 64
  const int wn = wave & 3;               // 0..3 : 32-col band of 128
  const int arow0 = (wm * 2 + 0) * 16 + lm;
  const int arow1 = (wm * 2 + 1) * 16 + lm;
  const int bcol0 = (wn * 2 + 0) * 16 + lm;
  const int bcol1 = (wn * 2 + 1) * 16 + lm;
  v8f acc00 = {}, acc01 = {}, acc10 = {}, acc11 = {};

  const int ar = tid >> 3, ac = (tid & 7) * 4;   // A (and B^T) staging coords
  const int bkp = tid >> 5, bnc = (tid & 31) * 4;  // B (NN) staging coords

  for (int k0 = 0; k0 < K; k0 += 32) {
    // stage A tile 64x32 -> As[m][k]
    {
      float4 a0 = *reinterpret_cast<const float4*>(A + (long long)(m0 + ar) * K + k0 + ac);
      float4 a1 = *reinterpret_cast<const float4*>(A + (long long)(m0 + ar + 32) * K + k0 + ac);
      *reinterpret_cast<float4*>(&As[ar][ac]) = a0;
      *reinterpret_cast<float4*>(&As[ar + 32][ac]) = a1;
    }
    // stage B tile 32(k) x 128(n) -> Bs[k/2][n] (K-paired)
    if (!TRANSB) {
#pragma unroll
      for (int j = 0; j < 2; ++j) {
        const int kp = bkp + j * 8;  // 0..15
        float4 r0 = *reinterpret_cast<const float4*>(B + (long long)(k0 + 2 * kp) * N + n0 + bnc);
        float4 r1 = *reinterpret_cast<const float4*>(B + (long long)(k0 + 2 * kp + 1) * N + n0 + bnc);
        Bs[kp][bnc + 0] = make_float2(r0.x, r1.x);
        Bs[kp][bnc + 1] = make_float2(r0.y, r1.y);
        Bs[kp][bnc + 2] = make_float2(r0.z, r1.z);
        Bs[kp][bnc + 3] = make_float2(r0.w, r1.w);
      }
    } else {
#pragma unroll
      for (int j = 0; j < 4; ++j) {
        const int n = ar + j * 32;  // 0..127
        float4 v = *reinterpret_cast<const float4*>(B + (long long)(n0 + n) * K + k0 + ac);
        Bs[(ac >> 1) + 0][n] = make_float2(v.x, v.y);
        Bs[(ac >> 1) + 1][n] = make_float2(v.z, v.w);
      }
    }
    __syncthreads();
#pragma unroll
    for (int kk = 0; kk < 32; kk += 4) {
      const int ks = kk + half * 2;  // half-wave holds K pair {ks, ks+1}
      const int kp = ks >> 1;
      v2f a0 = *reinterpret_cast<const v2f*>(&As[arow0][ks]);
      v2f a1 = *reinterpret_cast<const v2f*>(&As[arow1][ks]);
      v2f b0 = *reinterpret_cast<const v2f*>(&Bs[kp][bcol0]);
      v2f b1 = *reinterpret_cast<const v2f*>(&Bs[kp][bcol1]);
      acc00 = __builtin_amdgcn_wmma_f32_16x16x4_f32(false, a0, false, b0, (short)0, acc00,
                                                    false, false);
      acc01 = __builtin_amdgcn_wmma_f32_16x16x4_f32(false, a0, false, b1, (short)0, acc01,
                                                    false, false);
      acc10 = __builtin_amdgcn_wmma_f32_16x16x4_f32(false, a1, false, b0, (short)0, acc10,
                                                    false, false);
      acc11 = __builtin_amdgcn_wmma_f32_16x16x4_f32(false, a1, false, b1, (short)0, acc11,
                                                    false, false);
    }
    __syncthreads();
  }

  // C layout: VGPR v -> M = v + 8*half ; N = lane&15
  const int col0 = n0 + bcol0;
  const int col1 = n0 + bcol1;
  const float eb0 = (EPI == 2) ? E[col0] : 0.0f;
  const float eb1 = (EPI == 2) ? E[col1] : 0.0f;
#pragma unroll
  for (int i = 0; i < 2; ++i) {
    const int rbase = m0 + (wm * 2 + i) * 16 + half * 8;
    v8f accA = i ? acc10 : acc00;
    v8f accB = i ? acc11 : acc01;
#pragma unroll
    for (int v = 0; v < 8; ++v) {
      const int row = rbase + v;
      float e0 = accA[v], e1 = accB[v];
      if (EPI == 1) {
        e0 += E[(long long)row * N + col0];
        e1 += E[(long long)row * N + col1];
      } else if (EPI == 2) {
        e0 = fmaxf(e0 + eb0, 0.0f);
        e1 = fmaxf(e1 + eb1, 0.0f);
      }
      C[(long long)row * N + col0] = e0;
      C[(long long)row * N + col1] = e1;
    }
  }
}

// ---- bh[o] = vb_bh[o,:] . (n*exp(v_bh)) + mu_bh[o]  (one block per sample) ----
__global__ __launch_bounds__(256) void bh_kernel(float* __restrict__ bhb, long long sOut,
                                                 const float* __restrict__ vb,
                                                 const float* __restrict__ vbias,
                                                 const float* __restrict__ mub,
                                                 int s0, int tensorId) {
  __shared__ float rb[512];
  const int tid = threadIdx.x;
  const int s = blockIdx.x;
  uint32_t r[4];
  philox4x32((uint32_t)tid, (uint32_t)(s0 + s), (uint32_t)tensorId, 0x5EED5EEDu,
             (uint32_t)(s0 + s), 0xB1A5F00Du ^ (uint32_t)tensorId, r);
  float n0, n1;
  boxmuller(r[0], r[1], n0, n1);
  rb[2 * tid]     = n0 * __expf(vbias[2 * tid]);
  rb[2 * tid + 1] = n1 * __expf(vbias[2 * tid + 1]);
  __syncthreads();
  float* o = bhb + (long long)s * sOut;
#pragma unroll
  for (int rep = 0; rep < 2; ++rep) {
    const int oi = tid + rep * 256;
    float acc = mub[oi];
    const float* row = vb + (long long)oi * 512;
    for (int k = 0; k < 512; k += 4)
      acc += row[k] * rb[k] + row[k + 1] * rb[k + 1] + row[k + 2] * rb[k + 2] +
             row[k + 3] * rb[k + 3];
    o[oi] = acc;
  }
}

// ---- output layer, one block per sample (tiny: ~1 MFLOP/sample) ------------
__global__ __launch_bounds__(256) void out_kernel(const float* __restrict__ h, long long sH,
                                                  const float* __restrict__ mu_Wo,
                                                  const float* __restrict__ v_Wo,
                                                  const float* __restrict__ vo_Wo,
                                                  const float* __restrict__ vi_Wo,
                                                  const float* __restrict__ mu_bo,
                                                  const float* __restrict__ v_bo,
                                                  const float* __restrict__ vb_bo,
                                                  float* __restrict__ y, int s0) {
  __shared__ float rW[1024];   // rWo [2][512]
  __shared__ float t2[1024];   // vo_Wo @ rWo
  __shared__ float Wo[1024];   // [2][512]
  __shared__ float boS[2];
  const int tid = threadIdx.x;
  const int s = blockIdx.x;
  const int g = s0 + s;
  {
    uint32_t r[4];
    philox4x32((uint32_t)tid, (uint32_t)g, 6u, 0x5EED5EEDu, (uint32_t)g,
               0xB1A5F00Du ^ 6u, r);
    float n0, n1, n2, n3;
    boxmuller(r[0], r[1], n0, n1);
    boxmuller(r[2], r[3], n2, n3);
    const int i = tid * 4;
    rW[i + 0] = n0 * __expf(v_Wo[i + 0]);
    rW[i + 1] = n1 * __expf(v_Wo[i + 1]);
    rW[i + 2] = n2 * __expf(v_Wo[i + 2]);
    rW[i + 3] = n3 * __expf(v_Wo[i + 3]);
  }
  if (tid == 0) {
    uint32_t r[4];
    philox4x32(0u, (uint32_t)g, 7u, 0x5EED5EEDu, (uint32_t)g, 0xB1A5F00Du ^ 7u, r);
    float n0, n1;
    boxmuller(r[0], r[1], n0, n1);
    const float r0 = n0 * __expf(v_bo[0]), r1 = n1 * __expf(v_bo[1]);
    boS[0] = vb_bo[0] * r0 + vb_bo[1] * r1 + mu_bo[0];
    boS[1] = vb_bo[2] * r0 + vb_bo[3] * r1 + mu_bo[1];
  }
  __syncthreads();
#pragma unroll
  for (int rep = 0; rep < 4; ++rep) {  // t2[c,k] = vo_Wo[c,0]*r[0,k] + vo_Wo[c,1]*r[1,k]
    const int i = tid + rep * 256;
    const int cc = i >> 9, k = i & 511;
    t2[i] = vo_Wo[cc * 2 + 0] * rW[k] + vo_Wo[cc * 2 + 1] * rW[512 + k];
  }
  __syncthreads();
  for (int rep = 0; rep < 4; ++rep) {  // Wo[c,i] = t2[c,:] . vi_Wo[i,:] + mu_Wo[c,i]
    const int idx = tid + rep * 256;
    const int cc = idx >> 9, ii = idx & 511;
    float acc = mu_Wo[idx];
    const float* vr = vi_Wo + (long long)ii * 512;
    const float* tr = t2 + cc * 512;
    for (int k = 0; k < 512; k += 4)
      acc += tr[k] * vr[k] + tr[k + 1] * vr[k + 1] + tr[k + 2] * vr[k + 2] +
             tr[k + 3] * vr[k + 3];
    Wo[idx] = acc;
  }
  __syncthreads();
  const float* hs = h + (long long)s * sH;
  for (int rep = 0; rep < 4; ++rep) {  // y[b,c] = h[b,:] . Wo[c,:] + bo[c]
    const int idx = tid + rep * 256;
    const int b = idx >> 1, cc = idx & 1;
    float acc = boS[cc];
    const float* hr = hs + (long long)b * 512;
    const float* wr = Wo + cc * 512;
    for (int k = 0; k < 512; k += 4)
      acc += hr[k] * wr[k] + hr[k + 1] * wr[k + 1] + hr[k + 2] * wr[k + 2] +
             hr[k + 3] * wr[k + 3];
    y[(long long)s * 1024 + idx] = acc;
  }
}

// ---------------------------------------------------------------------------
extern "C" void kernel_launch(void* const* d_in, const int* in_sizes, int n_in,
                              void* d_out, int out_size, void* d_ws, size_t ws_size,
                              hipStream_t stream) {
  const float* x     = (const float*)d_in[0];
  const float* mu_Wh = (const float*)d_in[1];
  const float* v_Wh  = (const float*)d_in[2];
  const float* vo_Wh = (const float*)d_in[3];
  const float* vi_Wh = (const float*)d_in[4];
  const float* mu_bh = (const float*)d_in[5];
  const float* v_bh  = (const float*)d_in[6];
  const float* vb_bh = (const float*)d_in[7];
  const float* mu_Wo = (const float*)d_in[8];
  const float* v_Wo  = (const float*)d_in[9];
  const float* vo_Wo = (const float*)d_in[10];
  const float* vi_Wo = (const float*)d_in[11];
  const float* mu_bo = (const float*)d_in[12];
  const float* v_bo  = (const float*)d_in[13];
  const float* vb_bo = (const float*)d_in[14];
  float* out = (float*)d_out;

  const int NS = 150;                    // reference ns (scalar input, fixed)
  const int M = 512 * 512;               // 262144 floats per 512x512 matrix
  // per-sample scratch (floats): h0, h1, tmp, rw(->Wh), bh(512)
  const long long PS = 4LL * M + 512;
  int sb = (int)(ws_size / ((size_t)PS * 4));
  if (sb < 1) sb = 1;
  if (sb > NS) sb = NS;

  float* ws  = (float*)d_ws;
  float* h0  = ws;
  float* h1  = ws + M;
  float* tmp = ws + 2LL * M;
  float* rw  = ws + 3LL * M;          // holds r, then overwritten with Wh
  float* bhb = ws + 4LL * M;

  for (int s0 = 0; s0 < NS; s0 += sb) {
    const int cnt = (NS - s0 < sb) ? (NS - s0) : sb;
    for (int l = 0; l < 3; ++l) {
      // r = normal * exp(v_Wh[l])
      rng_fill<<<dim3(256, cnt), 256, 0, stream>>>(rw, PS, v_Wh + (long long)l * M, s0, l);
      // bh = vb_bh[l] @ rbh + mu_bh[l]
      bh_kernel<<<cnt, 256, 0, stream>>>(bhb, PS, vb_bh + (long long)l * M,
                                         v_bh + l * 512, mu_bh + l * 512, s0, 3 + l);
      // tmp = vo_Wh[l] @ r            (NN)
      gemm512<false, 0><<<dim3(4, 8, cnt), 256, 0, stream>>>(
          vo_Wh + (long long)l * M, 0, rw, PS, nullptr, 0, tmp, PS);
      // Wh = tmp @ vi_Wh[l]^T + mu_Wh[l]   (NT, write over r buffer)
      gemm512<true, 1><<<dim3(4, 8, cnt), 256, 0, stream>>>(
          tmp, PS, vi_Wh + (long long)l * M, 0, mu_Wh + (long long)l * M, 0, rw, PS);
      // h' = relu(h @ Wh^T + bh)      (NT + bias + relu)
      const float* hin = (l == 0) ? x : ((l == 1) ? h1 : h0);
      const long long sHin = (l == 0) ? 0 : PS;   // x shared by all samples
      float* hout = (l == 1) ? h0 : h1;
      gemm512<true, 2><<<dim3(4, 8, cnt), 256, 0, stream>>>(
          hin, sHin, rw, PS, bhb, PS, hout, PS);
    }
    // output layer (tiny) + final GEMV, one block per sample
    out_kernel<<<cnt, 256, 0, stream>>>(h1, PS, mu_Wo, v_Wo, vo_Wo, vi_Wo, mu_bo,
                                        v_bo, vb_bo, out + (long long)s0 * 1024, s0);
  }
}